// TableTransform_72782515798800
// MI455X (gfx1250) — compile-verified
//
#include <hip/hip_runtime.h>

// ---------------------------------------------------------------------------
// Problem constants (from reference): N=16384, S=64, O=64, G=32, H=32, NH=16,
// FS=128.  Memory-bound (~90MB @ 23.3TB/s); all matmuls in f32 WMMA
// (v_wmma_f32_16x16x4_f32), bulk staging via GLOBAL_LOAD_ASYNC_TO_LDS.
// ---------------------------------------------------------------------------
#define NROWS 16384
#define SDIM  64
#define ODIM  64
#define GDIM  32
#define HDIM  32
#define NHDIM 16
#define FSDIM 128
#define RB    16      // n-rows per workgroup
#define NTHREADS 256
#define NWAVES 8
#define NEGV  (-1e30f)

typedef __attribute__((ext_vector_type(2))) float v2f;
typedef __attribute__((ext_vector_type(8))) float v8f;

// D = A(16x4) * B(4x16) + C(16x16), all f32, native CDNA5 WMMA.
__device__ __forceinline__ v8f wmma4(v2f a, v2f b, v8f c) {
#if defined(__gfx1250__) && __has_builtin(__builtin_amdgcn_wmma_f32_16x16x4_f32)
  // 8 args: (neg_a, A, neg_b, B, c_mod, C, reuse_a, reuse_b)
  return __builtin_amdgcn_wmma_f32_16x16x4_f32(
      false, a, false, b, (short)0, c, false, false);
#else
  // Host pass / builtin-missing placeholder (never used on gfx1250 device).
  c[0] += a.x * b.x + a.y * b.y;
  return c;
#endif
}

// Async global->LDS staging (ASYNCcnt path), with sync fallback.
#if defined(__gfx1250__) &&                                          \
    __has_builtin(__builtin_amdgcn_global_load_async_to_lds_b128) && \
    __has_builtin(__builtin_amdgcn_global_load_async_to_lds_b32)
#define ASYNC_LDS 1
typedef __attribute__((ext_vector_type(4))) int i32x4;
#define GAS __attribute__((address_space(1)))
#define LAS __attribute__((address_space(3)))
__device__ __forceinline__ void async_cp_b128(void* lds, const void* g) {
  __builtin_amdgcn_global_load_async_to_lds_b128((GAS i32x4*)g, (LAS i32x4*)lds,
                                                 0, 0);
}
__device__ __forceinline__ void async_cp_b32(void* lds, const void* g) {
  __builtin_amdgcn_global_load_async_to_lds_b32((GAS int*)g, (LAS int*)lds,
                                                0, 0);
}
__device__ __forceinline__ void async_wait_all() {
  asm volatile("s_wait_asynccnt 0x0" ::: "memory");
}
#endif

__global__ __launch_bounds__(NTHREADS) void fused_tabletransform(
    const float* __restrict__ tfil, const float* __restrict__ tglb,
    const float* __restrict__ toh,  const float* __restrict__ toth,
    const float* __restrict__ Wot,  const float* __restrict__ Wemb,
    const float* __restrict__ Wprep,const float* __restrict__ Wsch,
    const float* __restrict__ bsch, const float* __restrict__ Wefc,
    const float* __restrict__ Wl,   const float* __restrict__ Wr,
    const float* __restrict__ Wtail,const int* __restrict__ maskp,
    const int* __restrict__ edgep,  float* __restrict__ outp) {
  // ---- LDS staging (~213 KB of the 320 KB/WGP) ----
  __shared__ float s_oh[RB * 96];              // [onehot | global] rows
  __shared__ float s_fL[RB * SDIM];            // -log(clip(1 - f*m))
  __shared__ float s_fR[RB * SDIM];            // -log(clip(1 - (1-f)*m))
  __shared__ float s_eF[RB * SDIM];            // edge as float
  __shared__ float s_mF[RB * SDIM];            // mask as float
  __shared__ float s_others[RB * SDIM * 13];
  __shared__ float s_emb[RB * HDIM * HDIM];    // per-n 32x32 embedding matrix
  __shared__ float s_wprep[16 * HDIM];         // 13x32 zero-padded to 16x32
  __shared__ float s_wsch[HDIM * 48];
  __shared__ float s_bsch[64];
  __shared__ float s_wsmall[3 * NHDIM * HDIM]; // Wl, Wr, Wefc
  __shared__ float s_part[3 * RB * 4 * NHDIM]; // per-(kind,n,s-tile) max partials
  __shared__ float s_acc[3 * RB * NHDIM];      // finalized left/right/edge
  __shared__ float s_res[RB * 224];            // concat result feeding tail gemm
  __shared__ float s_scr[NWAVES * 2 * 16 * 32];// per-wave C-tile relayout scratch

  const int tid  = threadIdx.x;
  const int lane = tid & 31;
  const int wv   = tid >> 5;
  const int hi   = lane >> 4;      // half-wave select (WMMA frag layout)
  const int lm   = lane & 15;
  const int n0   = blockIdx.x * RB;

  // ---- Phase 0: stage inputs & small weights ----
#ifdef ASYNC_LDS
  {  // bulk copies: VGPR-free async DMA into LDS (ASYNCcnt)
    const float* src = toth + (size_t)n0 * SDIM * 13;
    for (int i = tid; i < RB * SDIM * 13 / 4; i += NTHREADS)  // 13 exact iters
      async_cp_b128(&s_others[i * 4], &src[i * 4]);
    for (int i = tid; i < RB * 96; i += NTHREADS) {           // 6 exact iters
      int n = i / 96, c = i % 96;
      const float* g = (c < ODIM) ? &toh[(size_t)(n0 + n) * ODIM + c]
                                  : &tglb[(size_t)(n0 + n) * GDIM + (c - ODIM)];
      async_cp_b32(&s_oh[i], g);
    }
  }
#else
  for (int i = tid; i < RB * 96; i += NTHREADS) {
    int n = i / 96, c = i % 96;
    s_oh[i] = (c < ODIM) ? toh[(size_t)(n0 + n) * ODIM + c]
                         : tglb[(size_t)(n0 + n) * GDIM + (c - ODIM)];
  }
  for (int i = tid; i < RB * SDIM * 13; i += NTHREADS)
    s_others[i] = toth[(size_t)n0 * SDIM * 13 + i];
#endif
  for (int i = tid; i < RB * SDIM; i += NTHREADS) {
    int n = i / SDIM, s = i % SDIM;
    float f = tfil[(size_t)(n0 + n) * SDIM + s];
    float m = (float)maskp[(size_t)(n0 + n) * SDIM + s];
    float e = (float)edgep[(size_t)(n0 + n) * SDIM + s];
    s_fL[i] = -__logf(fmaxf(1.0f - f * m, 1e-9f));
    s_fR[i] = -__logf(fmaxf(1.0f - (1.0f - f) * m, 1e-9f));
    s_eF[i] = e;
    s_mF[i] = m;
  }
  for (int i = tid; i < 16 * HDIM; i += NTHREADS) {
    int r = i / HDIM, c = i % HDIM;
    s_wprep[i] = (r < 13) ? Wprep[r * HDIM + c] : 0.0f;
  }
  for (int i = tid; i < HDIM * 48; i += NTHREADS) s_wsch[i] = Wsch[i];
  for (int i = tid; i < 64; i += NTHREADS) s_bsch[i] = (i < 48) ? bsch[i] : 0.0f;
  for (int i = tid; i < NHDIM * HDIM; i += NTHREADS) {
    s_wsmall[0 * NHDIM * HDIM + i] = Wl[i];
    s_wsmall[1 * NHDIM * HDIM + i] = Wr[i];
    s_wsmall[2 * NHDIM * HDIM + i] = Wefc[i];
  }
  // Speculative L2 prefetch of the next block's table_others tile.
  if (blockIdx.x + 1 < gridDim.x) {
    const char* nxt = (const char*)(toth + (size_t)(n0 + RB) * SDIM * 13);
    for (int i = tid; i < 416; i += NTHREADS)       // 416 x 128B lines
      __builtin_prefetch(nxt + i * 128, 0, 1);
  }
#ifdef ASYNC_LDS
  async_wait_all();
#endif
  __syncthreads();

  // ---- Phase 1: emb = onehot(16x64) @ W_emb(64x1024) ----
  for (int t = wv; t < 64; t += NWAVES) {
    v8f c = {};
    for (int kk = 0; kk < 16; ++kk) {
      int k = 4 * kk + 2 * hi;
      v2f a; a.x = s_oh[lm * 96 + k]; a.y = s_oh[lm * 96 + k + 1];
      v2f b; b.x = Wemb[(size_t)k * 1024 + 16 * t + lm];
             b.y = Wemb[(size_t)(k + 1) * 1024 + 16 * t + lm];
      c = wmma4(a, b, c);
    }
    for (int v = 0; v < 8; ++v)
      s_emb[(v + 8 * hi) * 1024 + 16 * t + lm] = c[v];
  }
  __syncthreads();

  // ---- Phase 2: fused per-(n, s-tile) chain + masked-max partials ----
  float* scrA = &s_scr[wv * 2 * 512];
  float* scrB = scrA + 512;
  for (int job = wv; job < RB * 4; job += NWAVES) {
    int n = job >> 2, mt = job & 3;
    const float* oth = &s_others[(n * SDIM + 16 * mt) * 13];
    // others(16x13) @ W_prep -> scrA (16x32)
    for (int nt = 0; nt < 2; ++nt) {
      v8f c = {};
      for (int kk = 0; kk < 4; ++kk) {
        int k = 4 * kk + 2 * hi;
        v2f a;
        a.x = (k     < 13) ? oth[lm * 13 + k]     : 0.0f;
        a.y = (k + 1 < 13) ? oth[lm * 13 + k + 1] : 0.0f;
        v2f b; b.x = s_wprep[k * HDIM + 16 * nt + lm];
               b.y = s_wprep[(k + 1) * HDIM + 16 * nt + lm];
        c = wmma4(a, b, c);
      }
      for (int v = 0; v < 8; ++v) scrA[(v + 8 * hi) * 32 + 16 * nt + lm] = c[v];
    }
    // scrA(16x32) @ emb_n(32x32), relu -> scrB
    const float* embn = &s_emb[n * 1024];
    for (int nt = 0; nt < 2; ++nt) {
      v8f c = {};
      for (int kk = 0; kk < 8; ++kk) {
        int k = 4 * kk + 2 * hi;
        v2f a; a.x = scrA[lm * 32 + k]; a.y = scrA[lm * 32 + k + 1];
        v2f b; b.x = embn[k * 32 + 16 * nt + lm];
               b.y = embn[(k + 1) * 32 + 16 * nt + lm];
        c = wmma4(a, b, c);
      }
      for (int v = 0; v < 8; ++v)
        scrB[(v + 8 * hi) * 32 + 16 * nt + lm] = fmaxf(c[v], 0.0f);
    }
    // scrB(16x32) @ W_schema(32x48) + b, then masked-max partials
    for (int nt = 0; nt < 3; ++nt) {
      v8f c = {};
      for (int kk = 0; kk < 8; ++kk) {
        int k = 4 * kk + 2 * hi;
        v2f a; a.x = scrB[lm * 32 + k]; a.y = scrB[lm * 32 + k + 1];
        v2f b; b.x = s_wsch[k * 48 + 16 * nt + lm];
               b.y = s_wsch[(k + 1) * 48 + 16 * nt + lm];
        c = wmma4(a, b, c);
      }
      float bias = s_bsch[16 * nt + lm];
      const float* fArr = (nt == 0) ? s_fL : (nt == 1) ? s_fR : s_eF;
      const float* mArr = (nt == 2) ? s_eF : s_mF;
      float mval = NEGV;
      for (int v = 0; v < 8; ++v) {
        int s = 16 * mt + v + 8 * hi;
        float f = fArr[n * SDIM + s];
        float m = mArr[n * SDIM + s];
        float cand = (m != 0.0f) ? (c[v] + bias) * f : NEGV;
        mval = fmaxf(mval, cand);
      }
      mval = fmaxf(mval, __shfl_xor(mval, 16, 32));
      if (hi == 0)
        s_part[((nt * RB + n) * 4 + mt) * NHDIM + lm] = mval;
    }
  }
  __syncthreads();

  // ---- Phase 3a: combine partials; empty rows -> 0 (masked_max semantics) ----
  for (int i = tid; i < 3 * RB * NHDIM; i += NTHREADS) {
    int g = i / NHDIM, j = i % NHDIM;
    float m = NEGV;
    for (int mt = 0; mt < 4; ++mt)
      m = fmaxf(m, s_part[(g * 4 + mt) * NHDIM + j]);
    s_acc[i] = (m <= -1e29f) ? 0.0f : m;
  }
  __syncthreads();

  // ---- Phase 3b: oh_trans gemm + {left,right,edge} gemms -> s_res ----
  for (int t = wv; t < 8; t += NWAVES) {
    v8f c = {};
    for (int kk = 0; kk < 24; ++kk) {
      int k = 4 * kk + 2 * hi;
      v2f a; a.x = s_oh[lm * 96 + k]; a.y = s_oh[lm * 96 + k + 1];
      v2f b; b.x = Wot[(size_t)k * FSDIM + 16 * t + lm];
             b.y = Wot[(size_t)(k + 1) * FSDIM + 16 * t + lm];
      c = wmma4(a, b, c);
    }
    for (int v = 0; v < 8; ++v)
      s_res[(v + 8 * hi) * 224 + 96 + 16 * t + lm] = c[v];
  }
  for (int job = wv; job < 6; job += NWAVES) {
    int gi = job >> 1, nt = job & 1;
    const float* A = &s_acc[gi * RB * NHDIM];
    const float* B = &s_wsmall[gi * NHDIM * HDIM];
    v8f c = {};
    for (int kk = 0; kk < 4; ++kk) {
      int k = 4 * kk + 2 * hi;
      v2f a; a.x = A[lm * NHDIM + k]; a.y = A[lm * NHDIM + k + 1];
      v2f b; b.x = B[k * HDIM + 16 * nt + lm];
             b.y = B[(k + 1) * HDIM + 16 * nt + lm];
      c = wmma4(a, b, c);
    }
    for (int v = 0; v < 8; ++v)
      s_res[(v + 8 * hi) * 224 + gi * 32 + 16 * nt + lm] = c[v];
  }
  __syncthreads();

  // ---- Phase 4: out = res(16x224) @ W_tail(224x128) -> HBM ----
  for (int t = wv; t < 8; t += NWAVES) {
    v8f c = {};
    for (int kk = 0; kk < 56; ++kk) {
      int k = 4 * kk + 2 * hi;
      v2f a; a.x = s_res[lm * 224 + k]; a.y = s_res[lm * 224 + k + 1];
      v2f b; b.x = Wtail[(size_t)k * FSDIM + 16 * t + lm];
             b.y = Wtail[(size_t)(k + 1) * FSDIM + 16 * t + lm];
      c = wmma4(a, b, c);
    }
    for (int v = 0; v < 8; ++v)
      outp[(size_t)(n0 + v + 8 * hi) * FSDIM + 16 * t + lm] = c[v];
  }
}

extern "C" void kernel_launch(void* const* d_in, const int* in_sizes, int n_in,
                              void* d_out, int out_size, void* d_ws, size_t ws_size,
                              hipStream_t stream) {
  (void)in_sizes; (void)n_in; (void)d_ws; (void)ws_size; (void)out_size;
  const float* tfil  = (const float*)d_in[0];
  const float* tglb  = (const float*)d_in[1];
  const float* toh   = (const float*)d_in[2];
  const float* toth  = (const float*)d_in[3];
  const float* Wot   = (const float*)d_in[4];
  const float* Wemb  = (const float*)d_in[5];
  const float* Wprep = (const float*)d_in[6];
  const float* Wsch  = (const float*)d_in[7];
  const float* bsch  = (const float*)d_in[8];
  const float* Wefc  = (const float*)d_in[9];
  const float* Wl    = (const float*)d_in[10];
  const float* Wr    = (const float*)d_in[11];
  const float* Wtail = (const float*)d_in[12];
  const int*   maskp = (const int*)d_in[13];
  const int*   edgep = (const int*)d_in[14];
  float* outp = (float*)d_out;

  dim3 grid(NROWS / RB), block(NTHREADS);
  hipLaunchKernelGGL(fused_tabletransform, grid, block, 0, stream,
                     tfil, tglb, toh, toth, Wot, Wemb, Wprep, Wsch, bsch,
                     Wefc, Wl, Wr, Wtail, maskp, edgep, outp);
}